// HungarianMatcher_317827580355
// MI455X (gfx1250) — compile-verified
//
#include <hip/hip_runtime.h>
#include <math.h>

// HungarianMatcher cost matrix: C[b,q,n] = 5*L1(pred_box, tgt_box) - softmax(logits)[tgt_label]
// Flat view: rows r = b*Q+q in [0,9600), cols n in [0,6400).
// Class cost done as one-hot GEMM on v_wmma_f32_16x16x4_f32 (exact in f32).

typedef __attribute__((ext_vector_type(2))) float v2f;
typedef __attribute__((ext_vector_type(8))) float v8f;

#define NCLS   92
#define NROWS  9600   // B*Q = 32*300
#define NCOLS  6400   // total targets
#define TILES_PER_BLOCK 8   // 8 waves, one 16x16 tile each

__global__ __launch_bounds__(256)
void hungarian_cost_kernel(const float* __restrict__ logits,   // [9600][92]
                           const float* __restrict__ pboxes,   // [9600][4]
                           const int*   __restrict__ tlabels,  // [6400]
                           const float* __restrict__ tboxes,   // [6400][4]
                           float* __restrict__ out)            // [9600][6400]
{
    __shared__ float ldsP[16][NCLS];   // logits tile -> softmax probs (in place)
    __shared__ float ldsPB[16 * 4];    // pred boxes for 16 rows
    __shared__ float ldsTB[128 * 4];   // tgt boxes for 128 cols
    __shared__ int   ldsLab[128];      // tgt labels for 128 cols

    const int tid = threadIdx.x;
    const int r0  = blockIdx.y * 16;                 // row-tile base  (0..9584)
    const int nb  = blockIdx.x * (16 * TILES_PER_BLOCK); // col base (0..6272)

    // ---- stage inputs into LDS (all L2-resident; read-shared across blocks) ----
    for (int idx = tid; idx < 16 * NCLS; idx += 256) {
        int i = idx / NCLS, c = idx - i * NCLS;
        ldsP[i][c] = logits[(size_t)(r0 + i) * NCLS + c];
    }
    if (tid < 64)  ldsPB[tid]  = pboxes[(size_t)r0 * 4 + tid];
    if (tid < 128) ldsLab[tid] = tlabels[nb + tid];
    for (int idx = tid; idx < 512; idx += 256)
        ldsTB[idx] = tboxes[(size_t)nb * 4 + idx];
    __syncthreads();

    // ---- in-LDS row softmax: wave w handles rows 2w, 2w+1 (wave32 shuffles) ----
    const int wave = tid >> 5;
    const int lane = tid & 31;
    for (int ri = 0; ri < 2; ++ri) {
        const int m = wave * 2 + ri;
        float x0 = ldsP[m][lane];
        float x1 = ldsP[m][lane + 32];
        const bool v2 = (lane < NCLS - 64);          // lane < 28
        float x2 = v2 ? ldsP[m][lane + 64] : -INFINITY;
        float mx = fmaxf(fmaxf(x0, x1), x2);
        for (int off = 16; off > 0; off >>= 1)
            mx = fmaxf(mx, __shfl_xor(mx, off, 32));
        float e0 = expf(x0 - mx);
        float e1 = expf(x1 - mx);
        float e2 = v2 ? expf(x2 - mx) : 0.0f;
        float s = e0 + e1 + e2;
        for (int off = 16; off > 0; off >>= 1)
            s += __shfl_xor(s, off, 32);
        float inv = 1.0f / s;
        ldsP[m][lane]      = e0 * inv;
        ldsP[m][lane + 32] = e1 * inv;
        if (v2) ldsP[m][lane + 64] = e2 * inv;
    }
    __syncthreads();

    // ---- per-wave 16x16 tile: class cost via one-hot WMMA, bbox L1 via VALU ----
    const int half = lane >> 4;          // 0: lanes 0-15, 1: lanes 16-31
    const int nl   = lane & 15;
    const int lcol = (wave << 4) + nl;   // local column 0..127
    const int lab  = ldsLab[lcol];
    const float tb0 = ldsTB[lcol * 4 + 0];
    const float tb1 = ldsTB[lcol * 4 + 1];
    const float tb2 = ldsTB[lcol * 4 + 2];
    const float tb3 = ldsTB[lcol * 4 + 3];

    v8f acc = {};   // accumulates P @ OneHot  -> prob[r, label[n]]
    // A 16x4 f32 layout: VGPR0 = K=(kb+2*half), VGPR1 = K=(kb+2*half+1), M = nl.
    // B 4x16 mirrors A: same K per lane-half, N = nl. One-hot built in registers.
    for (int kb = 0; kb < NCLS; kb += 4) {
        const int k0 = kb + (half << 1);
        v2f a;
        a.x = ldsP[nl][k0];
        a.y = ldsP[nl][k0 + 1];
        v2f bfrag;
        bfrag.x = (lab == k0)     ? 1.0f : 0.0f;
        bfrag.y = (lab == k0 + 1) ? 1.0f : 0.0f;
        acc = __builtin_amdgcn_wmma_f32_16x16x4_f32(
            /*neg_a=*/false, a, /*neg_b=*/false, bfrag,
            /*c_mod=*/(short)0, acc, /*reuse_a=*/false, /*reuse_b=*/false);
    }

    // C/D layout: VGPR v holds M = v + 8*half, N = nl.
    const int col = nb + lcol;
#pragma unroll
    for (int v = 0; v < 8; ++v) {
        const int M = v + (half << 3);
        const float l1 = fabsf(ldsPB[M * 4 + 0] - tb0)
                       + fabsf(ldsPB[M * 4 + 1] - tb1)
                       + fabsf(ldsPB[M * 4 + 2] - tb2)
                       + fabsf(ldsPB[M * 4 + 3] - tb3);
        out[(size_t)(r0 + M) * NCOLS + col] = 5.0f * l1 - acc[v];
    }
}

extern "C" void kernel_launch(void* const* d_in, const int* in_sizes, int n_in,
                              void* d_out, int out_size, void* d_ws, size_t ws_size,
                              hipStream_t stream) {
    (void)in_sizes; (void)n_in; (void)out_size; (void)d_ws; (void)ws_size;
    const float* logits  = (const float*)d_in[0];   // [32,300,92]
    const float* pboxes  = (const float*)d_in[1];   // [32,300,4]
    const int*   tlabels = (const int*)  d_in[2];   // [6400]
    const float* tboxes  = (const float*)d_in[3];   // [6400,4]
    float* out = (float*)d_out;                     // [32,300,6400]

    dim3 grid(NCOLS / (16 * TILES_PER_BLOCK), NROWS / 16);  // (50, 600)
    hungarian_cost_kernel<<<grid, 256, 0, stream>>>(logits, pboxes, tlabels, tboxes, out);
}